// HistogramEqualization_61229053772359
// MI455X (gfx1250) — compile-verified
//
#include <hip/hip_runtime.h>
#include <hip/hip_bf16.h>
#include <stdint.h>

#define NBINS   256
#define TILE_F  4096   // floats per TDM tile (16 KB)
#define TILE8   2048   // 8-byte units per tile (data_size = 8B)

typedef float        v2f   __attribute__((ext_vector_type(2)));
typedef float        v8f   __attribute__((ext_vector_type(8)));
typedef unsigned int u32x4 __attribute__((ext_vector_type(4)));
typedef int          i32x4 __attribute__((ext_vector_type(4)));
typedef int          i32x8 __attribute__((ext_vector_type(8)));

// ws layout (uint words): [0]=min key, [1]=max key, [2..257]=hist (u32), [258..513]=cdf (f32)

__device__ __forceinline__ unsigned enc_f32(float f) {
    unsigned b = __float_as_uint(f);
    return (b & 0x80000000u) ? ~b : (b | 0x80000000u);   // order-preserving key
}
__device__ __forceinline__ float dec_f32(unsigned k) {
    unsigned b = (k & 0x80000000u) ? (k & 0x7FFFFFFFu) : ~k;
    return __uint_as_float(b);
}

// ---------------------------------------------------------------- init ------
__global__ void he_init(unsigned* ws) {
    int t = threadIdx.x;
    if (t == 0) { ws[0] = 0xFFFFFFFFu; ws[1] = 0u; }
    if (t < NBINS) ws[2 + t] = 0u;
}

// ------------------------------------------------------------- min/max ------
__global__ void he_minmax(const float* __restrict__ x, long long n, unsigned* ws) {
    __shared__ float smn[256];
    __shared__ float smx[256];
    float mn =  3.4e38f, mx = -3.4e38f;
    const long long n4     = n >> 2;
    const long long stride = (long long)gridDim.x * blockDim.x;
    const float4* x4 = (const float4*)x;
    for (long long i = (long long)blockIdx.x * blockDim.x + threadIdx.x; i < n4; i += stride) {
        float4 v = x4[i];
        mn = fminf(mn, fminf(fminf(v.x, v.y), fminf(v.z, v.w)));
        mx = fmaxf(mx, fmaxf(fmaxf(v.x, v.y), fmaxf(v.z, v.w)));
    }
    for (long long i = n4 * 4 + (long long)blockIdx.x * blockDim.x + threadIdx.x; i < n; i += stride) {
        float v = x[i]; mn = fminf(mn, v); mx = fmaxf(mx, v);
    }
    smn[threadIdx.x] = mn; smx[threadIdx.x] = mx;
    __syncthreads();
    for (int s = blockDim.x >> 1; s > 0; s >>= 1) {
        if ((int)threadIdx.x < s) {
            smn[threadIdx.x] = fminf(smn[threadIdx.x], smn[threadIdx.x + s]);
            smx[threadIdx.x] = fmaxf(smx[threadIdx.x], smx[threadIdx.x + s]);
        }
        __syncthreads();
    }
    if (threadIdx.x == 0) {
        atomicMin(&ws[0], enc_f32(smn[0]));
        atomicMax(&ws[1], enc_f32(smx[0]));
    }
}

// ------------------------------------------------ TDM tile load (16 KB) -----
__device__ __forceinline__ void tdm_load_tile(const float* gsrc, float* ldst) {
    unsigned lds = (unsigned)(uintptr_t)ldst;                 // low 32b = LDS byte addr
    unsigned long long ga = (unsigned long long)(uintptr_t)gsrc;
    u32x4 g0;
    g0[0] = 1u;                                               // count=1 (valid), user mode
    g0[1] = lds;                                              // lds_addr
    g0[2] = (unsigned)(ga & 0xFFFFFFFFull);                   // global_addr[31:0]
    g0[3] = (unsigned)((ga >> 32) & 0x01FFFFFFull)            // global_addr[56:32]
          | (2u << 30);                                       // type=2 ("image")
    i32x8 g1;
    g1[0] = (3 << 16);                                        // data_size=3 (8B); no pad/iter/mask
    g1[1] = (int)(((unsigned)TILE8 & 0xFFFFu) << 16);         // tensor_dim0[15:0] -> bits[63:48]
    g1[2] = (int)((((unsigned)TILE8 >> 16) & 0xFFFFu)         // tensor_dim0[31:16]
          | (1u << 16));                                      // tensor_dim1 = 1
    g1[3] = (int)((unsigned)TILE8 << 16);                     // tile_dim0 = TILE8
    g1[4] = 1;                                                // tile_dim1 = 1, tile_dim2 = 0
    g1[5] = TILE8;                                            // tensor_dim0_stride[31:0]
    g1[6] = 0;
    g1[7] = 0;
    i32x4 gz4  = {0, 0, 0, 0};                                // groups 2/3 unused (<=2D)
    i32x8 gz8  = {0, 0, 0, 0, 0, 0, 0, 0};
    // 6-arg variant (this toolchain): (u32x4, i32x8, i32x4, i32x4, i32x8, i32 cpol)
    __builtin_amdgcn_tensor_load_to_lds(g0, g1, gz4, gz4, gz8, 0);
}

// ------------------------------------------- histogram (TDM -> LDS bins) ----
__global__ void he_hist(const float* __restrict__ x, long long n, unsigned* ws) {
    __shared__ __align__(16) float buf[2][TILE_F];
    __shared__ unsigned shist[NBINS];
    const int tid = threadIdx.x;
    if (tid < NBINS) shist[tid] = 0u;

    const float lo    = dec_f32(ws[0]);
    const float hi    = dec_f32(ws[1]);
    const float scale = (float)NBINS / (hi - lo);
    unsigned* ghist = ws + 2;

    const long long ntiles = n / TILE_F;

    // prologue: wave 0 kicks off first DMA
    if (tid < 32) {
        if ((long long)blockIdx.x < ntiles)
            tdm_load_tile(x + (long long)blockIdx.x * TILE_F, &buf[0][0]);
    }
    __syncthreads();   // shist zeroing visible

    int cur = 0;
    for (long long tt = blockIdx.x; tt < ntiles; tt += gridDim.x) {
        long long nxt = tt + gridDim.x;
        if (tid < 32) {                         // wave-0-uniform branch
            if (nxt < ntiles) {
                tdm_load_tile(x + nxt * TILE_F, &buf[cur ^ 1][0]);
                __builtin_amdgcn_s_wait_tensorcnt(1);   // tile tt is done
            } else {
                __builtin_amdgcn_s_wait_tensorcnt(0);
            }
        }
        __syncthreads();                        // publish buf[cur] to all waves
        for (int i = tid; i < TILE_F; i += 256) {
            float v = buf[cur][i];
            int b = (int)((v - lo) * scale);
            b = min(max(b, 0), NBINS - 1);
            atomicAdd(&shist[b], 1u);
        }
        __syncthreads();                        // all reads done before buffer reuse
        cur ^= 1;
    }
    // tail (n not a multiple of TILE_F) — block 0 only
    if (blockIdx.x == 0) {
        for (long long i = ntiles * TILE_F + tid; i < n; i += 256) {
            float v = x[i];
            int b = (int)((v - lo) * scale);
            b = min(max(b, 0), NBINS - 1);
            atomicAdd(&shist[b], 1u);
        }
    }
    __syncthreads();
    if (tid < NBINS) atomicAdd(&ghist[tid], shist[tid]);
}

// ------------------------------- 256-bin cumsum via WMMA (1 wave, EXEC=~0) --
// Scan-as-matmul: H[16][16], RP = H x U (U upper-triangular ones) via four
// accumulating V_WMMA_F32_16X16X4_F32, then add strictly-lower row offsets.
__global__ void he_scan(unsigned* ws) {
    __shared__ float h[NBINS];
    __shared__ float rp[NBINS];
    const int lane = threadIdx.x;              // 0..31, single wave
    const unsigned* ghist = ws + 2;
    for (int i = lane; i < NBINS; i += 32) h[i] = (float)ghist[i];
    __syncthreads();

    const int m  = lane & 15;                  // row (A) / col (B,C)
    const int kb = (lane < 16) ? 0 : 2;        // K-half select per ISA f32 layouts
    v8f c = {};
#pragma unroll
    for (int k = 0; k < 4; ++k) {
        v2f a, b;
        a[0] = h[16 * m + 4 * k + kb];         // A: 16x4 slice of H (cols 4k..4k+3)
        a[1] = h[16 * m + 4 * k + kb + 1];
        int r0 = 4 * k + kb;                   // B: rows 4k..4k+3 of U
        b[0] = (r0     <= m) ? 1.0f : 0.0f;
        b[1] = (r0 + 1 <= m) ? 1.0f : 0.0f;
        c = __builtin_amdgcn_wmma_f32_16x16x4_f32(
                false, a, false, b, (short)0, c, false, false);
    }
    const int rbase = (lane < 16) ? 0 : 8;     // C/D layout: VGPR j -> row j(+8)
#pragma unroll
    for (int j = 0; j < 8; ++j) rp[(j + rbase) * 16 + m] = c[j];
    __syncthreads();

    float rowtot[16];
#pragma unroll
    for (int r = 0; r < 16; ++r) rowtot[r] = rp[r * 16 + 15];
    float* cdf = (float*)(ws + 2 + NBINS);
#pragma unroll
    for (int j = 0; j < 8; ++j) {
        int r = j + rbase;
        float off = 0.0f, g = 0.0f;
        for (int r2 = 0; r2 < 16; ++r2) {
            if (r2 < r) off += rowtot[r2];
            g += rowtot[r2];
        }
        cdf[r * 16 + m] = (c[j] + off) / g;    // normalized inclusive CDF
    }
}

// ------------------------------------------------------- interp/map pass ----
__device__ __forceinline__ float he_lut(float v, float lo, float invw, const float* cdf) {
    float t = fmaxf((v - lo) * invw - 0.5f, 0.0f);
    t = fminf(t, 255.0f);
    int j = min((int)t, 254);
    float f = t - (float)j;
    float c0 = cdf[j], c1 = cdf[j + 1];
    return c0 + (c1 - c0) * f;                 // == np.interp on uniform centers
}

__global__ void he_map(const float* __restrict__ x, float* __restrict__ out,
                       long long n, const unsigned* __restrict__ ws) {
    __shared__ float scdf[NBINS];
    if (threadIdx.x < NBINS)
        scdf[threadIdx.x] = ((const float*)(ws + 2 + NBINS))[threadIdx.x];
    __syncthreads();
    const float lo   = dec_f32(ws[0]);
    const float hi   = dec_f32(ws[1]);
    const float invw = (float)NBINS / (hi - lo);

    const long long n4     = n >> 2;
    const long long stride = (long long)gridDim.x * blockDim.x;
    const float4* x4 = (const float4*)x;
    float4*       o4 = (float4*)out;
    for (long long i = (long long)blockIdx.x * blockDim.x + threadIdx.x; i < n4; i += stride) {
        float4 v = x4[i];
        float4 r;
        r.x = he_lut(v.x, lo, invw, scdf);
        r.y = he_lut(v.y, lo, invw, scdf);
        r.z = he_lut(v.z, lo, invw, scdf);
        r.w = he_lut(v.w, lo, invw, scdf);
        o4[i] = r;
    }
    for (long long i = n4 * 4 + (long long)blockIdx.x * blockDim.x + threadIdx.x; i < n; i += stride)
        out[i] = he_lut(x[i], lo, invw, scdf);
}

// ----------------------------------------------------------------------------
extern "C" void kernel_launch(void* const* d_in, const int* in_sizes, int n_in,
                              void* d_out, int out_size, void* d_ws, size_t ws_size,
                              hipStream_t stream) {
    const float* x  = (const float*)d_in[0];
    float*       out = (float*)d_out;
    long long    n  = (long long)in_sizes[0];
    unsigned*    ws = (unsigned*)d_ws;

    he_init<<<1, 256, 0, stream>>>(ws);
    he_minmax<<<2048, 256, 0, stream>>>(x, n, ws);

    long long ntiles = n / TILE_F;
    long long hb = ntiles < 1 ? 1 : (ntiles < 2048 ? ntiles : 2048);
    he_hist<<<(int)hb, 256, 0, stream>>>(x, n, ws);

    he_scan<<<1, 32, 0, stream>>>(ws);
    he_map<<<4096, 256, 0, stream>>>(x, out, n, ws);
}